// ChamferLoss_17592186045168
// MI455X (gfx1250) — compile-verified
//
#include <hip/hip_runtime.h>
#include <hip/hip_bf16.h>
#include <math.h>

// ---------------------------------------------------------------------------
// ChamferLoss top-K kernel for MI455X (gfx1250, wave32).
//
// d2[m][n] = ||q_n||^2 + ||r_m||^2 - 2 q_n . r_m  is produced directly by one
// V_WMMA_F32_16X16X4_F32 per 16x16 tile:
//   A row m (16x4):  (-2 r.x, -2 r.y, -2 r.z, ||r||^2)
//   B col n (4x16):  ( q.x,    q.y,    q.z,   1      )
//   C[m][n]       :   ||q_n||^2
//
// Top-4 selection is wave-uniformly skipped for tiles where no lane can
// improve its current 4th-best (ballot -> scalar branch, EXEC stays full).
// ---------------------------------------------------------------------------

typedef __attribute__((ext_vector_type(2))) float v2f;
typedef __attribute__((ext_vector_type(8))) float v8f;

#define BATCH 4
#define DIM   3
#define KNN   4

__global__ void chamfer_init_kernel(float* out) {
    if (threadIdx.x == 0 && blockIdx.x == 0) out[0] = 0.0f;
}

__device__ __forceinline__ void top4_insert8(const v8f& d,
                                             float& t0, float& t1,
                                             float& t2, float& t3) {
#pragma unroll
    for (int i = 0; i < 8; ++i) {
        const float dv  = d[i];
        const bool lt3 = dv < t3;
        const bool lt2 = dv < t2;
        const bool lt1 = dv < t1;
        const bool lt0 = dv < t0;
        t3 = lt3 ? (lt2 ? t2 : dv) : t3;
        t2 = lt2 ? (lt1 ? t1 : dv) : t2;
        t1 = lt1 ? (lt0 ? t0 : dv) : t1;
        t0 = lt0 ? dv : t0;
    }
}

__global__ __launch_bounds__(128) void chamfer_topk_kernel(
    const float* __restrict__ query,   // [B, N, 3]
    const float* __restrict__ ref,     // [B, M, 3]
    float* __restrict__ out,           // [1]
    int N, int M, float inv_total)
{
    const int lane        = threadIdx.x & 31;
    const int waveInBlock = threadIdx.x >> 5;
    const int tile        = blockIdx.x * (blockDim.x >> 5) + waveInBlock;

    const int tilesPerBatch = N >> 4;               // N / 16
    const int b     = tile / tilesPerBatch;
    const int qtile = tile - b * tilesPerBatch;

    const int  col = lane & 15;                     // query column this lane owns
    const bool hi  = lane >= 16;
    const int  n   = qtile * 16 + col;              // query index within batch

    // ---- Load this lane's query point, build B operand and C (=||q||^2) ----
    const float* qp = query + ((size_t)b * N + n) * DIM;
    const float qx = qp[0], qy = qp[1], qz = qp[2];
    const float qq = qx * qx + qy * qy + qz * qz;

    // B (4x16, K x N): lanes 0-15 hold K=0 (v0) and K=1 (v1);
    //                  lanes 16-31 hold K=2 (v0) and K=3 (v1).
    v2f bmat;
    bmat.x = hi ? qz   : qx;
    bmat.y = hi ? 1.0f : qy;

    v8f c;
#pragma unroll
    for (int i = 0; i < 8; ++i) c[i] = qq;          // C[m][n] = ||q_n||^2

    // ---- Per-lane sorted top-4 (ascending) over this lane's ref rows ----
    float t0 = INFINITY, t1 = INFINITY, t2 = INFINITY, t3 = INFINITY;

    const float* rbase = ref + (size_t)b * M * DIM;

#pragma unroll 2
    for (int m0 = 0; m0 < M; m0 += 16) {
        const int    m  = m0 + col;
        const float* rp = rbase + (size_t)m * DIM;
        const float rx = rp[0], ry = rp[1], rz = rp[2];
        const float rr = rx * rx + ry * ry + rz * rz;

        // A (16x4, M x K): lanes 0-15 -> K=0 (v0), K=1 (v1);
        //                  lanes 16-31 -> K=2 (v0), K=3 (v1).
        v2f a;
        a.x = hi ? (-2.0f * rz) : (-2.0f * rx);
        a.y = hi ? rr           : (-2.0f * ry);

        // d2 tile: lane holds query col `col`, ref rows m0 + i + (hi ? 8 : 0)
        v8f d = __builtin_amdgcn_wmma_f32_16x16x4_f32(
            /*neg_a=*/false, a, /*neg_b=*/false, bmat,
            /*c_mod=*/(short)0, c, /*reuse_a=*/false, /*reuse_b=*/false);

        // Cheap wave-uniform filter: skip the insert network when no lane in
        // the wave can improve its current 4th-best distance. The ballot
        // lowers to an SGPR compare + s_cbranch: no divergence, EXEC stays
        // all-ones (WMMA above is outside the branch).
        const float dmin =
            fminf(fminf(fminf(d[0], d[1]), fminf(d[2], d[3])),
                  fminf(fminf(d[4], d[5]), fminf(d[6], d[7])));

        if (__any(dmin < t3)) {
            top4_insert8(d, t0, t1, t2, t3);
        }
    }

    // ---- Merge sorted top-4 with partner lane (same query, other ref rows) ----
    const float o0 = __shfl_xor(t0, 16, 32);
    const float o1 = __shfl_xor(t1, 16, 32);
    const float o2 = __shfl_xor(t2, 16, 32);
    const float o3 = __shfl_xor(t3, 16, 32);

    // k-th smallest of two sorted 4-lists: min(a_k, b_k, min_{i+j=k-1} max(a_i, b_j))
    const float m0v = fminf(t0, o0);
    const float m1v = fminf(fmaxf(t0, o0), fminf(t1, o1));
    const float m2v = fminf(fminf(t2, o2),
                            fminf(fmaxf(t0, o1), fmaxf(t1, o0)));
    const float m3v = fminf(fminf(t3, o3),
                            fminf(fmaxf(t1, o1),
                                  fminf(fmaxf(t0, o2), fmaxf(t2, o0))));

    float s = hi ? 0.0f : (m0v + m1v + m2v + m3v);  // count each query once

    // ---- Wave reduction, then one atomic per wave ----
#pragma unroll
    for (int off = 16; off > 0; off >>= 1)
        s += __shfl_down(s, off, 32);

    if (lane == 0)
        atomicAdd(out, s * inv_total);
}

extern "C" void kernel_launch(void* const* d_in, const int* in_sizes, int n_in,
                              void* d_out, int out_size, void* d_ws, size_t ws_size,
                              hipStream_t stream) {
    (void)n_in; (void)d_ws; (void)ws_size; (void)out_size;

    const float* query = (const float*)d_in[0];
    const float* ref   = (const float*)d_in[1];
    float*       out   = (float*)d_out;

    const int N = in_sizes[0] / (BATCH * DIM);      // 8192
    const int M = in_sizes[1] / (BATCH * DIM);      // 8192

    const float inv_total = 1.0f / ((float)BATCH * (float)N * (float)KNN);

    chamfer_init_kernel<<<1, 32, 0, stream>>>(out);

    const int tiles          = BATCH * (N / 16);    // 2048 wave-tiles
    const int wavesPerBlock  = 4;                   // 128 threads
    const int blocks         = tiles / wavesPerBlock;

    chamfer_topk_kernel<<<blocks, wavesPerBlock * 32, 0, stream>>>(
        query, ref, out, N, M, inv_total);
}